// DeepseekV2ForCausalLM_50835232916125
// MI455X (gfx1250) — compile-verified
//
#include <hip/hip_runtime.h>
#include <hip/hip_bf16.h>
#include <math.h>

// Problem constants (match reference)
#define T_TOK 256
#define H_DIM 2048
#define E_EXP 64
#define I_DIM 1408
#define TOPK  6

#define KC    32     // K-chunk staged through LDS
#define NTILE 128    // output-column tile per block (8 waves x 16 cols)
#define NPAD  4      // LDS row padding (reproduced by TDM pad feature)

typedef __attribute__((ext_vector_type(2))) float v2f;
typedef __attribute__((ext_vector_type(8))) float v8f;
typedef unsigned int __attribute__((ext_vector_type(4))) u32x4;
typedef int          __attribute__((ext_vector_type(4))) i32x4;
typedef int          __attribute__((ext_vector_type(8))) i32x8;

#if __has_builtin(__builtin_amdgcn_tensor_load_to_lds)
#define HAVE_TDM 1
#else
#define HAVE_TDM 0
#endif

// ---------------------------------------------------------------------------
// Tensor Data Mover: load a KC x NTILE f32 tile from a row-major matrix
// (row length = row_len elements) into LDS at byte offset lds_off, padding
// NPAD dwords after every NTILE dwords (pad_interval=128dw -> code 6,
// pad_amount=4dw -> code 3).  D# packing per CDNA5 ISA sec. 8.3/8.4.
// ---------------------------------------------------------------------------
#if HAVE_TDM
__device__ __forceinline__ void tdm_issue_tile(const float* gsrc, unsigned lds_off,
                                               unsigned row_len, unsigned rows)
{
    unsigned long long ga = (unsigned long long)(uintptr_t)gsrc;
    u32x4 g0;
    g0.x = 1u;                                           // count=1 (user D#)
    g0.y = lds_off;                                      // lds_addr
    g0.z = (unsigned)ga;                                 // global_addr[31:0]
    g0.w = ((unsigned)(ga >> 32) & 0x01FFFFFFu)          // global_addr[56:32]
         | (2u << 30);                                   // type=2 ("image")
    i32x8 g1;
    g1[0] = (int)((2u << 16)                             // data_size=4B
                | (1u << 20)                             // pad_enable
                | (6u << 22)                             // pad_interval: 128 dw
                | (3u << 25));                           // pad_amount: 4 dw
    g1[1] = (int)((row_len & 0xFFFFu) << 16);            // tensor_dim0[15:0]
    g1[2] = (int)(((row_len >> 16) & 0xFFFFu)            // tensor_dim0[31:16]
                | ((rows & 0xFFFFu) << 16));             // tensor_dim1[15:0]
    g1[3] = (int)(((rows >> 16) & 0xFFFFu)               // tensor_dim1[31:16]
                | ((unsigned)NTILE << 16));              // tile_dim0
    g1[4] = (int)KC;                                     // tile_dim1 (tile_dim2=0)
    g1[5] = (int)row_len;                                // tensor_dim0_stride[31:0]
    g1[6] = 0;                                           // stride0 hi / stride1 lo
    g1[7] = 0;
    i32x4 gz = (i32x4)0;
#if defined(__clang_major__) && (__clang_major__ >= 23)
    i32x8 gz8 = (i32x8)0;
    __builtin_amdgcn_tensor_load_to_lds(g0, g1, gz, gz, gz8, 0);
#else
    __builtin_amdgcn_tensor_load_to_lds(g0, g1, gz, gz, 0);
#endif
}
#endif

// ---------------------------------------------------------------------------
// Router: logits -> top-6 of logits -> softmax over the 6 -> weights + lists
// ---------------------------------------------------------------------------
__global__ __launch_bounds__(64) void router_kernel(
    const float* __restrict__ x, const float* __restrict__ gw,
    float* __restrict__ topw, int* __restrict__ cnt, int* __restrict__ list)
{
    const int t = blockIdx.x;
    const int tid = threadIdx.x;          // expert id, 0..63
    const float* xr = x + (size_t)t * H_DIM;

    float acc = 0.f;
    for (int h = 0; h < H_DIM; ++h)
        acc = fmaf(xr[h], gw[(size_t)h * E_EXP + tid], acc);

    __shared__ float lg[E_EXP];
    lg[tid] = acc;
    __syncthreads();

    if (tid == 0) {
        int idx[TOPK];
        float val[TOPK];
        unsigned long long used = 0ull;
        #pragma unroll
        for (int k = 0; k < TOPK; ++k) {
            float best = -INFINITY; int bi = 0;
            for (int e = 0; e < E_EXP; ++e) {
                float v = lg[e];
                if (!((used >> e) & 1ull) && v > best) { best = v; bi = e; }
            }
            used |= (1ull << bi);
            idx[k] = bi; val[k] = best;
        }
        // softmax->topk->renorm == softmax over just the top-k logits
        float mx = val[0];
        float w[TOPK]; float sum = 0.f;
        #pragma unroll
        for (int k = 0; k < TOPK; ++k) { w[k] = __expf(val[k] - mx); sum += w[k]; }
        float inv = 1.f / sum;
        #pragma unroll
        for (int k = 0; k < TOPK; ++k) {
            int e = idx[k];
            topw[t * TOPK + k] = w[k] * inv;
            int slot = atomicAdd(&cnt[e], 1);
            list[e * T_TOK + slot] = (t << 3) | k;   // pack token + k-slot
        }
    }
}

// ---------------------------------------------------------------------------
// WMMA GEMM cores (templated on #16-row M tiles; accumulators in registers).
// Fragment layouts per CDNA5 ISA 7.12.2:
//   A (16x4 f32): lane%16 = M, K = j + 2*(lane>=16) for VGPR j
//   B (4x16 f32): K = j + 2*(lane>=16), N = lane%16
//   C/D (16x16) : VGPR r holds M=r (lanes 0-15) and M=r+8 (lanes 16-31)
// B-tiles double-buffered via TDM (TENSORcnt), issued by wave 0 only
// (scalar-branch guarded: TDM ignores EXEC, so a vector-predicated skip
// would still issue it from every wave).
// ---------------------------------------------------------------------------

template<int MT>
__device__ __forceinline__ void ffn1_core(
    int n, int i0,
    const float* __restrict__ x, const float* __restrict__ w1e,
    const float* __restrict__ topw, const int* __restrict__ listE,
    float* __restrict__ Hbuf,
    float (*xs)[KC + 1], float (*bs)[KC][NTILE + NPAD])
{
    const int tid  = threadIdx.x;
    const int lane = tid & 31;
    const int wave = tid >> 5;            // 8 waves -> 8 x 16 output columns
    const int ln   = lane & 15;
    const int hi   = lane >> 4;           // lane half
    const int wave_s = __builtin_amdgcn_readfirstlane(wave);

    v8f acc[MT];
    #pragma unroll
    for (int m = 0; m < MT; ++m) acc[m] = (v8f){0,0,0,0,0,0,0,0};

    const int NCH = H_DIM / KC;
#if HAVE_TDM
    if (wave_s == 0)
        tdm_issue_tile(w1e + i0, (unsigned)(uintptr_t)&bs[0][0][0], I_DIM, H_DIM);
#endif

    for (int j = 0; j < NCH; ++j) {
        const int k0  = j * KC;
        const int buf = j & 1;
#if !HAVE_TDM
        for (int i = tid; i < KC * NTILE; i += 256) {
            int r = i / NTILE, c = i % NTILE;
            bs[buf][r][c] = w1e[(size_t)(k0 + r) * I_DIM + i0 + c];
        }
#endif
        // gather X rows for this expert's tokens (zero-pad past n)
        for (int i = tid; i < MT * 16 * KC; i += 256) {
            int m = i / KC, c = i % KC;
            float v = 0.f;
            if (m < n) {
                int t = listE[m] >> 3;
                v = x[(size_t)t * H_DIM + k0 + c];
            }
            xs[m][c] = v;
        }
#if HAVE_TDM
        if (wave_s == 0) {
            if (j + 1 < NCH) {
                tdm_issue_tile(w1e + (size_t)(k0 + KC) * I_DIM + i0,
                               (unsigned)(uintptr_t)&bs[buf ^ 1][0][0], I_DIM, H_DIM);
                __builtin_amdgcn_s_wait_tensorcnt(1);   // chunk j landed
            } else {
                __builtin_amdgcn_s_wait_tensorcnt(0);
            }
        }
#endif
        __syncthreads();

        #pragma unroll
        for (int kk = 0; kk < KC / 4; ++kk) {
            v2f b;
            b.x = bs[buf][kk * 4 + 2 * hi    ][wave * 16 + ln];
            b.y = bs[buf][kk * 4 + 2 * hi + 1][wave * 16 + ln];
            #pragma unroll
            for (int m = 0; m < MT; ++m) {
                v2f a;
                a.x = xs[m * 16 + ln][kk * 4 + 2 * hi    ];
                a.y = xs[m * 16 + ln][kk * 4 + 2 * hi + 1];
                acc[m] = __builtin_amdgcn_wmma_f32_16x16x4_f32(
                    false, a, false, b, (short)0, acc[m], false, false);
            }
        }
        __syncthreads();
    }

    // epilogue: silu * routing weight -> Hbuf[pair][i]
    #pragma unroll
    for (int m = 0; m < MT; ++m) {
        #pragma unroll
        for (int r = 0; r < 8; ++r) {
            int row = m * 16 + r + 8 * hi;
            if (row < n) {
                int ent = listE[row];
                int t = ent >> 3, k = ent & 7;
                float v = acc[m][r];
                float s = v / (1.f + __expf(-v));       // silu
                s *= topw[t * TOPK + k];
                Hbuf[(size_t)(t * TOPK + k) * I_DIM + i0 + wave * 16 + ln] = s;
            }
        }
    }
}

template<int MT>
__device__ __forceinline__ void ffn2_core(
    int n, int h0,
    const float* __restrict__ Hbuf, const float* __restrict__ w2e,
    const int* __restrict__ listE, float* __restrict__ out,
    float (*xs)[KC + 1], float (*bs)[KC][NTILE + NPAD])
{
    const int tid  = threadIdx.x;
    const int lane = tid & 31;
    const int wave = tid >> 5;
    const int ln   = lane & 15;
    const int hi   = lane >> 4;
    const int wave_s = __builtin_amdgcn_readfirstlane(wave);

    v8f acc[MT];
    #pragma unroll
    for (int m = 0; m < MT; ++m) acc[m] = (v8f){0,0,0,0,0,0,0,0};

    const int NCH = I_DIM / KC;
#if HAVE_TDM
    if (wave_s == 0)
        tdm_issue_tile(w2e + h0, (unsigned)(uintptr_t)&bs[0][0][0], H_DIM, I_DIM);
#endif

    for (int j = 0; j < NCH; ++j) {
        const int k0  = j * KC;
        const int buf = j & 1;
#if !HAVE_TDM
        for (int i = tid; i < KC * NTILE; i += 256) {
            int r = i / NTILE, c = i % NTILE;
            bs[buf][r][c] = w2e[(size_t)(k0 + r) * H_DIM + h0 + c];
        }
#endif
        for (int i = tid; i < MT * 16 * KC; i += 256) {
            int m = i / KC, c = i % KC;
            float v = 0.f;
            if (m < n) {
                int ent = listE[m];
                int pair = (ent >> 3) * TOPK + (ent & 7);
                v = Hbuf[(size_t)pair * I_DIM + k0 + c];
            }
            xs[m][c] = v;
        }
#if HAVE_TDM
        if (wave_s == 0) {
            if (j + 1 < NCH) {
                tdm_issue_tile(w2e + (size_t)(k0 + KC) * H_DIM + h0,
                               (unsigned)(uintptr_t)&bs[buf ^ 1][0][0], H_DIM, I_DIM);
                __builtin_amdgcn_s_wait_tensorcnt(1);
            } else {
                __builtin_amdgcn_s_wait_tensorcnt(0);
            }
        }
#endif
        __syncthreads();

        #pragma unroll
        for (int kk = 0; kk < KC / 4; ++kk) {
            v2f b;
            b.x = bs[buf][kk * 4 + 2 * hi    ][wave * 16 + ln];
            b.y = bs[buf][kk * 4 + 2 * hi + 1][wave * 16 + ln];
            #pragma unroll
            for (int m = 0; m < MT; ++m) {
                v2f a;
                a.x = xs[m * 16 + ln][kk * 4 + 2 * hi    ];
                a.y = xs[m * 16 + ln][kk * 4 + 2 * hi + 1];
                acc[m] = __builtin_amdgcn_wmma_f32_16x16x4_f32(
                    false, a, false, b, (short)0, acc[m], false, false);
            }
        }
        __syncthreads();
    }

    #pragma unroll
    for (int m = 0; m < MT; ++m) {
        #pragma unroll
        for (int r = 0; r < 8; ++r) {
            int row = m * 16 + r + 8 * hi;
            if (row < n) {
                int t = listE[row] >> 3;
                atomicAdd(out + (size_t)t * H_DIM + h0 + wave * 16 + ln, acc[m][r]);
            }
        }
    }
}

// ---------------------------------------------------------------------------
// Kernels: one block per (expert, output-column tile); template-dispatch on
// the expert's M-tile count (n is uniform per block -> wave-uniform WMMA).
// ---------------------------------------------------------------------------
__global__ __launch_bounds__(256) void ffn1_kernel(
    const float* __restrict__ x, const float* __restrict__ w1,
    const float* __restrict__ topw, const int* __restrict__ cnt,
    const int* __restrict__ list, float* __restrict__ Hbuf)
{
    __shared__ float xs[16 * 16][KC + 1];
    __shared__ float bs[2][KC][NTILE + NPAD];
    const int e  = blockIdx.x;
    const int i0 = blockIdx.y * NTILE;
    const int n  = cnt[e];
    if (n == 0) return;
    const float* w1e   = w1 + (size_t)e * H_DIM * I_DIM;
    const int*   listE = list + e * T_TOK;
    const int nt = (n + 15) >> 4;
    if      (nt <= 1) ffn1_core<1 >(n, i0, x, w1e, topw, listE, Hbuf, xs, bs);
    else if (nt <= 2) ffn1_core<2 >(n, i0, x, w1e, topw, listE, Hbuf, xs, bs);
    else if (nt <= 4) ffn1_core<4 >(n, i0, x, w1e, topw, listE, Hbuf, xs, bs);
    else if (nt <= 8) ffn1_core<8 >(n, i0, x, w1e, topw, listE, Hbuf, xs, bs);
    else              ffn1_core<16>(n, i0, x, w1e, topw, listE, Hbuf, xs, bs);
}

__global__ __launch_bounds__(256) void ffn2_kernel(
    const float* __restrict__ Hbuf, const float* __restrict__ w2,
    const int* __restrict__ cnt, const int* __restrict__ list,
    float* __restrict__ out)
{
    __shared__ float xs[16 * 16][KC + 1];
    __shared__ float bs[2][KC][NTILE + NPAD];
    const int e  = blockIdx.x;
    const int h0 = blockIdx.y * NTILE;
    const int n  = cnt[e];
    if (n == 0) return;
    const float* w2e   = w2 + (size_t)e * I_DIM * H_DIM;
    const int*   listE = list + e * T_TOK;
    const int nt = (n + 15) >> 4;
    if      (nt <= 1) ffn2_core<1 >(n, h0, Hbuf, w2e, listE, out, xs, bs);
    else if (nt <= 2) ffn2_core<2 >(n, h0, Hbuf, w2e, listE, out, xs, bs);
    else if (nt <= 4) ffn2_core<4 >(n, h0, Hbuf, w2e, listE, out, xs, bs);
    else if (nt <= 8) ffn2_core<8 >(n, h0, Hbuf, w2e, listE, out, xs, bs);
    else              ffn2_core<16>(n, h0, Hbuf, w2e, listE, out, xs, bs);
}

// ---------------------------------------------------------------------------
// Launch
// ---------------------------------------------------------------------------
extern "C" void kernel_launch(void* const* d_in, const int* in_sizes, int n_in,
                              void* d_out, int out_size, void* d_ws, size_t ws_size,
                              hipStream_t stream)
{
    (void)in_sizes; (void)n_in; (void)ws_size;
    const float* x  = (const float*)d_in[0];
    const float* gw = (const float*)d_in[1];
    const float* w1 = (const float*)d_in[2];
    const float* w2 = (const float*)d_in[3];
    float* out = (float*)d_out;

    // workspace layout
    float* Hbuf = (float*)d_ws;                                   // [1536][I]
    float* topw = Hbuf + (size_t)(T_TOK * TOPK) * I_DIM;          // [1536]
    int*   cnt  = (int*)(topw + T_TOK * TOPK);                    // [64]
    int*   list = cnt + E_EXP;                                    // [64][256]

    hipMemsetAsync(cnt, 0, E_EXP * sizeof(int), stream);
    hipMemsetAsync(out, 0, (size_t)out_size * sizeof(float), stream);

    router_kernel<<<T_TOK, 64, 0, stream>>>(x, gw, topw, cnt, list);
    ffn1_kernel<<<dim3(E_EXP, I_DIM / NTILE), 256, 0, stream>>>(x, w1, topw, cnt, list, Hbuf);
    ffn2_kernel<<<dim3(E_EXP, H_DIM / NTILE), 256, 0, stream>>>(Hbuf, w2, cnt, list, out);
}